// EfficientGraphModulatedAttention_31928786878551
// MI455X (gfx1250) — compile-verified
//
#include <hip/hip_runtime.h>

// Problem constants (from reference setup_inputs)
#define B_    32
#define C_    256      // DIM
#define HW_   4096     // 64*64
#define NH_   4
#define C4_   64       // DIM/4 (hidden of graph_modulator)
#define S_    40       // num_select = int(4096*0.01)
#define PIX_  64       // pixels per block in k1
#define NEG_BIG (-3.402823466e38f)

typedef __attribute__((ext_vector_type(2))) float v2f;
typedef __attribute__((ext_vector_type(8))) float v8f;

// ---------------------------------------------------------------------------
// K1: fused importance + spatial modulator (two 1x1 convs as GEMMs).
//   mod[b,hw] = 0.7 * mean_h sigmoid(W2 . relu(W1 . x[b,:,hw] + b1) + b2)
//   imp[b,hw] = mean_c |x[b,c,hw]|
// GEMM1 ([64 pix,256] x [256,64]) runs on V_WMMA_F32_16X16X4_F32.
// Each wave owns two 16x16 output tiles sharing the same N-column group, so
// every B (weight) fragment is loaded from LDS once and feeds two WMMAs with
// independent accumulators (halves B-side DS traffic, hides WMMA latency).
// ---------------------------------------------------------------------------
__global__ __launch_bounds__(256) void k1_modulator(
    const float* __restrict__ x,  const float* __restrict__ w1,
    const float* __restrict__ b1, const float* __restrict__ w2,
    const float* __restrict__ b2, float* __restrict__ mod,
    float* __restrict__ imp)
{
    __shared__ float xs [C_ * PIX_];          // [k][m]  stride 64
    __shared__ float w1s[C_ * (C4_ + 1)];     // [k][n]  stride 65 (pad: no bank conflicts)
    __shared__ float h1s[PIX_ * (C4_ + 1)];   // [m][n]  stride 65

    const int tid = threadIdx.x;
    const int bb  = blockIdx.x >> 6;          // batch
    const int hw0 = (blockIdx.x & 63) * PIX_; // pixel tile base

    // ---- stage x tile [256 ch][64 contiguous pixels] (float4, coalesced) ----
    const float* xb = x + (size_t)bb * C_ * HW_ + hw0;
    for (int i = tid; i < C_ * (PIX_ / 4); i += 256) {
        const int c = i >> 4, j = (i & 15) * 4;
        float4 v = *(const float4*)(xb + (size_t)c * HW_ + j);
        *(float4*)(xs + c * PIX_ + j) = v;
    }
    // ---- stage W1 transposed: w1s[k][n] = w1[n][k] ----
    for (int i = tid; i < C4_ * C_; i += 256) {
        const int n = i >> 8, k = i & 255;
        w1s[k * (C4_ + 1) + n] = w1[n * C_ + k];
    }
    __syncthreads();

    // ---- importance = mean_c |x| (x already resident in LDS) ----
    if (tid < PIX_) {
        float s = 0.f;
        for (int c = 0; c < C_; ++c) s += fabsf(xs[c * PIX_ + tid]);
        imp[bb * HW_ + hw0 + tid] = s * (1.0f / (float)C_);
    }

    // ---- GEMM1: h1[64,64] = xs^T[64,256] . w1s[256,64], WMMA f32 16x16x4 ----
    const int wave = tid >> 5, lane = tid & 31;
    const int lhalf = lane >> 4, l15 = lane & 15;

    const int n0  = (wave & 3) * 16;      // this wave's N-column group
    const int m0a = (wave >> 2) * 16;     // tile A rows: 0 or 16
    const int m0b = m0a + 32;             // tile B rows: 32 or 48
    const int n   = n0 + l15;
    const int ma  = m0a + l15, mb = m0b + l15;

    v8f acc0 = {}, acc1 = {};
    for (int kk = 0; kk < C_ / 4; ++kk) {
        // ISA 7.12.2 fp32 A 16x4: lanes0-15 -> K{k0,k0+1}, lanes16-31 -> K{k0+2,k0+3}
        const int r0 = kk * 4 + (lhalf << 1);
        v2f bf, a0, a1;
        bf[0] = w1s[ r0      * (C4_ + 1) + n];
        bf[1] = w1s[(r0 + 1) * (C4_ + 1) + n];
        a0[0] = xs [ r0      * PIX_ + ma];
        a0[1] = xs [(r0 + 1) * PIX_ + ma];
        a1[0] = xs [ r0      * PIX_ + mb];
        a1[1] = xs [(r0 + 1) * PIX_ + mb];
        acc0 = __builtin_amdgcn_wmma_f32_16x16x4_f32(
            false, a0, false, bf, (short)0, acc0, false, false);
        acc1 = __builtin_amdgcn_wmma_f32_16x16x4_f32(
            false, a1, false, bf, (short)0, acc1, false, false);
    }
    // C/D layout: VGPR g -> row m0+g (lanes0-15) / m0+g+8 (lanes16-31), col n
    {
        const float bias = b1[n];
        for (int g = 0; g < 8; ++g) {
            const int Ma = m0a + g + (lhalf << 3);
            const int Mb = m0b + g + (lhalf << 3);
            h1s[Ma * (C4_ + 1) + n] = fmaxf(acc0[g] + bias, 0.f);   // ReLU
            h1s[Mb * (C4_ + 1) + n] = fmaxf(acc1[g] + bias, 0.f);
        }
    }
    __syncthreads();

    // ---- GEMM2 [64,64]x[64->4] + sigmoid + head mean (tiny, VALU) ----
    if (tid < PIX_) {
        float msum = 0.f;
        for (int h = 0; h < NH_; ++h) {
            float a = b2[h];
            for (int j = 0; j < C4_; ++j)
                a += h1s[tid * (C4_ + 1) + j] * w2[h * C4_ + j];
            msum += 1.0f / (1.0f + expf(-a));
        }
        mod[bb * HW_ + hw0 + tid] = 0.7f * (msum * (1.0f / (float)NH_));
    }
}

// ---------------------------------------------------------------------------
// K2: per-batch top-40 of importance; each selected pixel's modulation gets
// +0.3/S (the graph-attention branch provably yields pixel_imp == 1/S because
// attn.mean over the softmax axis of each head-row is exactly 1/S).
// Tie-break: lower index wins (matches jax.lax.top_k ordering).
// ---------------------------------------------------------------------------
__global__ __launch_bounds__(256) void k2_topk(const float* __restrict__ imp,
                                               float* __restrict__ mod)
{
    __shared__ float vals[HW_];
    __shared__ float rv[256];
    __shared__ int   ri[256];
    const int tid = threadIdx.x, b = blockIdx.x;

    for (int i = tid; i < HW_; i += 256) vals[i] = imp[b * HW_ + i];
    __syncthreads();

    for (int s = 0; s < S_; ++s) {
        float best = NEG_BIG; int bidx = HW_;
        for (int i = tid; i < HW_; i += 256) {       // i strictly increasing ->
            const float v = vals[i];                 // strict > keeps lowest idx
            if (v > best) { best = v; bidx = i; }
        }
        rv[tid] = best; ri[tid] = bidx;
        __syncthreads();
        for (int off = 128; off > 0; off >>= 1) {
            if (tid < off) {
                const float v2 = rv[tid + off]; const int i2 = ri[tid + off];
                if (v2 > rv[tid] || (v2 == rv[tid] && i2 < ri[tid])) {
                    rv[tid] = v2; ri[tid] = i2;
                }
            }
            __syncthreads();
        }
        if (tid == 0) {
            const int idx = ri[0];
            mod[b * HW_ + idx] += 0.3f / (float)S_;
            vals[idx] = NEG_BIG;
        }
        __syncthreads();
    }
}

// ---------------------------------------------------------------------------
// K3: out = x * mod (broadcast over channels). float4 path; mod map (512 KB)
// stays L2-resident so HBM traffic is just x-in + out-out.
// ---------------------------------------------------------------------------
__global__ __launch_bounds__(256) void k3_apply(const float* __restrict__ x,
                                                const float* __restrict__ mod,
                                                float* __restrict__ out)
{
    const size_t q    = (size_t)blockIdx.x * 256 + threadIdx.x;
    const size_t base = q * 4;                       // element index
    const int b  = (int)(base >> 20);                // C_*HW_ = 1<<20
    const int hw = (int)(base & (HW_ - 1));          // HW_ = 1<<12; 4096%4==0 ok
    const float4 xv = *(const float4*)(x + base);
    const float4 mv = *(const float4*)(mod + (size_t)b * HW_ + hw);
    float4 o;
    o.x = xv.x * mv.x; o.y = xv.y * mv.y;
    o.z = xv.z * mv.z; o.w = xv.w * mv.w;
    *(float4*)(out + base) = o;
}

// ---------------------------------------------------------------------------
extern "C" void kernel_launch(void* const* d_in, const int* in_sizes, int n_in,
                              void* d_out, int out_size, void* d_ws, size_t ws_size,
                              hipStream_t stream) {
    (void)in_sizes; (void)n_in; (void)out_size; (void)ws_size;
    const float* x  = (const float*)d_in[0];
    const float* w1 = (const float*)d_in[1];   // gm_w1 [64,256]
    const float* b1 = (const float*)d_in[2];   // gm_b1 [64]
    const float* w2 = (const float*)d_in[3];   // gm_w2 [4,64]
    const float* b2 = (const float*)d_in[4];   // gm_b2 [4]
    // qkv_w / graph encoder inputs are unused: the attention branch reduces
    // exactly to the constant 1/S per selected pixel (softmax rows sum to 1).

    float* mod = (float*)d_ws;                 // [B_*HW_] floats
    float* imp = mod + (size_t)B_ * HW_;       // [B_*HW_] floats
    float* out = (float*)d_out;

    k1_modulator<<<B_ * (HW_ / PIX_), 256, 0, stream>>>(x, w1, b1, w2, b2, mod, imp);
    k2_topk    <<<B_,                 256, 0, stream>>>(imp, mod);
    k3_apply   <<<(B_ * C_ * HW_ / 4) / 256, 256, 0, stream>>>(x, mod, out);
}